// MatrixTree_13408887898300
// MI455X (gfx1250) — compile-verified
//
#include <hip/hip_runtime.h>
#include <math.h>

// MatrixTree marginals: batched 1024x1024 fp32 inversion via blocked
// Gauss-Jordan on augmented [L^T | I], 32-wide pivot blocks, trailing
// updates with v_wmma_f32_16x16x4_f32 (fp32 WMMA: precision needed for
// an inverse; KB=32 halves HBM passes vs KB=16 -> ~13 GB total traffic).

#define EPSV 1e-5f
constexpr int BB   = 32;    // batch
constexpr int NN   = 1024;  // matrix dim
constexpr int NA   = 2048;  // augmented width
constexpr int KB   = 32;    // pivot block width
constexpr int NBLK = NN / KB;  // 32 steps

typedef __attribute__((ext_vector_type(2))) float v2f;
typedef __attribute__((ext_vector_type(8))) float v8f;

// ---------------------------------------------------------------- colsum
// colsum[b][c] = sum_{r<L, r!=c} (exp(S[b][r][c]) + EPS)
__global__ void mt_colsum(const float* __restrict__ S,
                          const int* __restrict__ len,
                          float* __restrict__ CS) {
  int b = blockIdx.y;
  int c = blockIdx.x * blockDim.x + threadIdx.x;
  int L = len[b];
  float s = 0.f;
  if (c < L) {
    const float* col = S + (size_t)b * NN * NN + c;
    for (int r = 0; r < L; ++r)
      if (r != c) s += expf(col[(size_t)r * NN]) + EPSV;
  }
  CS[b * NN + c] = s;
}

// ---------------------------------------------------------------- build AUG = [L^T | I]
__global__ void mt_build(const float* __restrict__ S,
                         const int* __restrict__ len,
                         const float* __restrict__ CS,
                         float* __restrict__ AUG) {
  int b  = blockIdx.z;
  int i0 = blockIdx.y * 16, j0 = blockIdx.x * 16;
  int tx = threadIdx.x, ty = threadIdx.y;   // tx -> j (coalesced), ty -> i
  int L  = len[b];
  __shared__ float st[16][17];              // st[jj][ii] = S[b][j][i]
  const float* Sb = S + (size_t)b * NN * NN;
  if (j0 < NN)
    st[ty][tx] = Sb[(size_t)(j0 + ty) * NN + (i0 + tx)];
  __syncthreads();
  int i = i0 + ty, j = j0 + tx;
  float v;
  if (j0 >= NN) {                            // right half: identity
    v = (i == (j - NN)) ? 1.f : 0.f;
  } else {
    bool vi = i < L, vj = j < L;
    if (!(vi && vj))      v = (i == j) ? 1.f : 0.f;         // pad -> identity
    else if (j == 0)      v = expf(Sb[(size_t)i * NN + i]); // root row of L
    else if (i == j)      v = CS[b * NN + i];               // diag(colsum)
    else                  v = -(expf(st[tx][ty]) + EPSV);   // -A (transposed)
  }
  AUG[((size_t)b * NN + i) * NA + j] = v;
}

// ---------------------------------------------------------------- invert 32x32 pivot block
// 64 threads (2 waves); lane owns one of 64 augmented columns.
__global__ void mt_pivinv(const float* __restrict__ AUG,
                          float* __restrict__ DINV, int k) {
  int b = blockIdx.x;
  int c = threadIdx.x;                       // 0..63
  __shared__ float W[KB][2 * KB + 1];
  const float* Dsrc = AUG + ((size_t)b * NN + (size_t)k * KB) * NA + (size_t)k * KB;
#pragma unroll
  for (int r = 0; r < KB; ++r)
    W[r][c] = (c < KB) ? Dsrc[(size_t)r * NA + c]
                       : ((c - KB == r) ? 1.f : 0.f);
  __syncthreads();
  for (int p = 0; p < KB; ++p) {
    float pr = W[p][c] * (1.f / W[p][p]);
    float colp[KB];
#pragma unroll
    for (int r = 0; r < KB; ++r) colp[r] = W[r][p];
    __syncthreads();
    W[p][c] = pr;
#pragma unroll
    for (int r = 0; r < KB; ++r)
      if (r != p) W[r][c] -= colp[r] * pr;
    __syncthreads();
  }
  if (c >= KB)
#pragma unroll
    for (int r = 0; r < KB; ++r)
      DINV[((size_t)b * KB + r) * KB + (c - KB)] = W[r][c];
}

// ---------------------------------------------------------------- pivot rows <- Dinv * rows
// One wave per 16-col tile; computes both 16-row halves of the 32-row panel.
// All B-operand slices are register-staged BEFORE any store (in-place panel).
__global__ void mt_scale(float* __restrict__ AUG,
                         const float* __restrict__ DINV, int k) {
  int b = blockIdx.y;
  int col0 = (2 * k + blockIdx.x) * 16;      // active tiles: cols [32k, 2048)
  int lane = threadIdx.x;
  int lh = lane & 15;
  int hi = lane >> 4;
  const float* Db = DINV + (size_t)b * KB * KB;       // 32x32 row-major
  const float* R  = AUG + ((size_t)b * NN + (size_t)k * KB) * NA + col0;
  // stage B: 8 K-slices of the 32xK panel tile (read all before writing any)
  v2f bm[8];
#pragma unroll
  for (int s = 0; s < 8; ++s) {
    int kb = s * 4 + 2 * hi;
    bm[s].x = R[(size_t)kb * NA + lh];
    bm[s].y = R[(size_t)(kb + 1) * NA + lh];
  }
  v8f accT = {}, accB = {};
#pragma unroll
  for (int s = 0; s < 8; ++s) {              // K=32 in slices of 4
    int kb = s * 4 + 2 * hi;
    v2f aT, aB;
    aT.x = Db[lh * KB + kb];                 // rows 0..15 of Dinv
    aT.y = Db[lh * KB + kb + 1];
    aB.x = Db[(16 + lh) * KB + kb];          // rows 16..31 of Dinv
    aB.y = Db[(16 + lh) * KB + kb + 1];
    accT = __builtin_amdgcn_wmma_f32_16x16x4_f32(false, aT, false, bm[s],
                                                 (short)0, accT, false, false);
    accB = __builtin_amdgcn_wmma_f32_16x16x4_f32(false, aB, false, bm[s],
                                                 (short)0, accB, false, false);
  }
  float* WpT = AUG + ((size_t)b * NN + (size_t)k * KB + 8 * hi) * NA + col0 + lh;
  float* WpB = WpT + (size_t)16 * NA;
#pragma unroll
  for (int r = 0; r < 8; ++r) {              // D: M = r + 8*hi, N = lh
    WpT[(size_t)r * NA] = accT[r];
    WpB[(size_t)r * NA] = accB[r];
  }
}

// ---------------------------------------------------------------- trailing update: M[i,:] -= C * P
__global__ void mt_update(float* __restrict__ AUG, int k) {
  int b = blockIdx.y;
  int iblk = blockIdx.x;                     // 16-row block
  if ((iblk >> 1) == k) return;              // pivot row blocks are done
  __shared__ float negC[16][KB + 1];         // 16 x 32 column block, negated
  int tid = threadIdx.x;                     // 256 threads = 8 waves
  {
    int r = tid >> 4, c2 = (tid & 15) * 2;   // coalesced float2 per row
    const float2 cc = *(const float2*)(AUG +
        ((size_t)b * NN + (size_t)iblk * 16 + r) * NA + (size_t)k * KB + c2);
    negC[r][c2]     = -cc.x;
    negC[r][c2 + 1] = -cc.y;
  }
  __syncthreads();
  int wave = tid >> 5, lane = tid & 31;
  int lh = lane & 15, hi = lane >> 4;
  int ntiles = 128 - 2 * k;
  const float* Prow = AUG + ((size_t)b * NN + (size_t)k * KB) * NA;
  for (int t = wave; t < ntiles; t += 8) {
    int col0 = (2 * k + t) * 16;
    float* Dp = AUG + ((size_t)b * NN + (size_t)iblk * 16 + 8 * hi) * NA + col0 + lh;
    v8f acc;
#pragma unroll
    for (int r = 0; r < 8; ++r) acc[r] = Dp[(size_t)r * NA];
#pragma unroll
    for (int s = 0; s < 8; ++s) {            // K=32 in slices of 4
      int kb = s * 4 + 2 * hi;
      v2f a, bmat;
      a.x = negC[lh][kb];                    // A = -C (F32 WMMA has no A-neg)
      a.y = negC[lh][kb + 1];
      bmat.x = Prow[(size_t)kb * NA + col0 + lh];
      bmat.y = Prow[(size_t)(kb + 1) * NA + col0 + lh];
      acc = __builtin_amdgcn_wmma_f32_16x16x4_f32(false, a, false, bmat,
                                                  (short)0, acc, false, false);
    }
#pragma unroll
    for (int r = 0; r < 8; ++r) Dp[(size_t)r * NA] = acc[r];
  }
}

// ---------------------------------------------------------------- finalize
// out[b,i,j] = (expS*invdiag_j*(j>0) - expS*invT[i][j]*(i>0) + (i==j)*expd*invT[0][i]) * v2d
__global__ void mt_finalize(const float* __restrict__ S,
                            const int* __restrict__ len,
                            const float* __restrict__ AUG,
                            float* __restrict__ out) {
  int b = blockIdx.z;
  int i = blockIdx.y * 16 + threadIdx.y;
  int j = blockIdx.x * 16 + threadIdx.x;
  int L = len[b];
  float v = 0.f;
  if (i < L && j < L) {
    const float* Sb   = S + (size_t)b * NN * NN;
    const float* INVT = AUG + (size_t)b * NN * NA + NN;  // right half = inv(L)^T
    float es = expf(Sb[(size_t)i * NN + j]);
    float t1 = (j == 0) ? 0.f : es * INVT[(size_t)j * NA + j]; // inv_diag[j]
    float t2 = (i == 0) ? 0.f : es * INVT[(size_t)i * NA + j]; // inv[j][i]
    v = t1 - t2;
    if (i == j) v += es * INVT[i];                             // expd_i * inv[i][0]
  }
  out[((size_t)b * NN + i) * NN + j] = v;
}

extern "C" void kernel_launch(void* const* d_in, const int* in_sizes, int n_in,
                              void* d_out, int out_size, void* d_ws, size_t ws_size,
                              hipStream_t stream) {
  (void)in_sizes; (void)n_in; (void)out_size; (void)ws_size;
  const float* S   = (const float*)d_in[0];
  const int*   len = (const int*)d_in[1];
  float* out  = (float*)d_out;
  float* AUG  = (float*)d_ws;                         // B*N*2N fp32 (256 MB)
  float* CS   = AUG + (size_t)BB * NN * NA;           // B*N
  float* DINV = CS + (size_t)BB * NN;                 // B*32*32

  mt_colsum<<<dim3(NN / 256, BB), 256, 0, stream>>>(S, len, CS);
  mt_build<<<dim3(NA / 16, NN / 16, BB), dim3(16, 16), 0, stream>>>(S, len, CS, AUG);
  for (int k = 0; k < NBLK; ++k) {
    mt_pivinv<<<dim3(BB), 2 * KB, 0, stream>>>(AUG, DINV, k);
    mt_scale<<<dim3(128 - 2 * k, BB), 32, 0, stream>>>(AUG, DINV, k);
    mt_update<<<dim3(NN / 16, BB), 256, 0, stream>>>(AUG, k);
  }
  mt_finalize<<<dim3(NN / 16, NN / 16, BB), dim3(16, 16), 0, stream>>>(S, len, AUG, out);
}